// SSelfAttention_72593537237170
// MI455X (gfx1250) — compile-verified
//
#include <hip/hip_runtime.h>
#include <math.h>

// ---------------- problem constants ----------------
#define BB 2
#define NN 512
#define TT 12
#define EE 128
#define HH 8
#define DD 16
#define SS (TT * HH)                   /* 96 (t,h) pairs per batch          */
#define SLICES (BB * SS)               /* 192 independent attention slices  */
#define MARKOFF (-100000000.0f)

typedef __attribute__((ext_vector_type(2))) float v2f;   // A/B of wmma f32 16x16x4
typedef __attribute__((ext_vector_type(8))) float v8f;   // C/D of 16x16 f32 tile
typedef __attribute__((ext_vector_type(4))) unsigned int v4u;  // TDM D# group0
typedef __attribute__((ext_vector_type(8))) int v8i;           // TDM D# group1
typedef __attribute__((ext_vector_type(4))) int v4i;           // TDM D# group2/3

// D = A(16x4) x B(4x16) + C, chained for K via successive calls.
// Layout (ISA 7.12.2):
//   A: lane L holds M=L%16, K = k0 + 2*(L/16) (+1 for 2nd vgpr)
//   B: lane L holds N=L%16, K = k0 + 2*(L/16) (+1 for 2nd vgpr)
//   C/D: vgpr r, lane L -> M = r + 8*(L/16), N = L%16
__device__ __forceinline__ v8f wmma4(v2f a, v2f b, v8f c) {
    return __builtin_amdgcn_wmma_f32_16x16x4_f32(false, a, false, b,
                                                 (short)0, c, false, false);
}

// ---------------- kernel 0: zero column sums ----------------
__global__ void zero_colsum_kernel(float* __restrict__ colsum) {
    int i = blockIdx.x * blockDim.x + threadIdx.x;   // SLICES*NN threads exactly
    colsum[i] = 0.0f;
}

// ---------------- kernel 1: q/k/v head projections ----------------
// qp/kp/vp[((b*T+t)*H+h)*N + n][16] = X @ W^T  (W is 16x16)
__global__ void proj_kernel(const float* __restrict__ values,
                            const float* __restrict__ keys,
                            const float* __restrict__ query,
                            const float* __restrict__ Wv,
                            const float* __restrict__ Wk,
                            const float* __restrict__ Wq,
                            float* __restrict__ qp,
                            float* __restrict__ kp,
                            float* __restrict__ vp) {
    const size_t per = (size_t)SLICES * NN * DD;              // 1,572,864
    size_t j = (size_t)blockIdx.x * blockDim.x + threadIdx.x; // 3*per threads
    int mat = (int)(j / per);
    size_t rem = j - (size_t)mat * per;
    int dp = (int)(rem & 15);
    size_t R = rem >> 4;                 // R = ((b*T+t)*H+h)*N + n
    int n = (int)(R % NN);
    int h = (int)((R / NN) % HH);
    int t = (int)((R / (NN * HH)) % TT);
    int b = (int)(R / ((size_t)NN * HH * TT));

    const float* src = (mat == 0) ? query : ((mat == 1) ? keys : values);
    const float* W   = (mat == 0) ? Wq    : ((mat == 1) ? Wk   : Wv);
    float* dst       = (mat == 0) ? qp    : ((mat == 1) ? kp   : vp);

    const float* x = src + ((((size_t)b * NN + n) * TT + t) * EE + h * DD);
    float acc = 0.0f;
#pragma unroll
    for (int d = 0; d < DD; ++d) acc += x[d] * W[dp * DD + d];
    dst[R * DD + dp] = acc;
}

// ========================================================================
// Shared core: one wave computes exp(clip(mask(QK^T)/scale)) for a 16x16
// (q,k) tile of one slice and accumulates per-column (per-k) sums.
// ========================================================================
__device__ __forceinline__ void energy_tile(const float* __restrict__ qp,
                                            const float* __restrict__ kp,
                                            const float* __restrict__ mask,
                                            float* __restrict__ colsum,
                                            int wid, int lane,
                                            int* slice_o, int* q0_o, int* k0_o,
                                            float p_out[8]) {
    const int kt = wid & 31;
    const int qt = (wid >> 5) & 31;
    const int slice = wid >> 10;                 // < 192, exact grid
    const int half = lane >> 4;
    const int lm = lane & 15;

    const float* Q = qp + (size_t)slice * NN * DD;
    const float* K = kp + (size_t)slice * NN * DD;
    const int q0 = qt * 16, k0 = kt * 16;

    v8f c = {};
#pragma unroll
    for (int kk = 0; kk < DD; kk += 4) {
        const int d0 = kk + 2 * half;
        v2f a, bm;
        a.x  = Q[(q0 + lm) * DD + d0];
        a.y  = Q[(q0 + lm) * DD + d0 + 1];
        bm.x = K[(k0 + lm) * DD + d0];       // B[d][kcol] = K[k0+kcol][d]
        bm.y = K[(k0 + lm) * DD + d0 + 1];
        c = wmma4(a, bm, c);
    }

    const float inv_scale = 0.08838834764831845f;   // 1/sqrt(128)
    const int kg = k0 + lm;
    float psum = 0.0f;
#pragma unroll
    for (int r = 0; r < 8; ++r) {
        const int qg = q0 + r + 8 * half;
        const float m = mask[qg * NN + kg];
        const float e = c[r] * m + MARKOFF * (1.0f - m);
        const float x = fminf(fmaxf(e * inv_scale, -5.0f), 5.0f);
        const float p = __expf(x);
        p_out[r] = p;
        psum += p;
    }
    psum += __shfl_down(psum, 16);               // lanes L / L+16: same column
    if (half == 0) atomicAdd(&colsum[(size_t)slice * NN + kg], psum);
    *slice_o = slice; *q0_o = q0; *k0_o = k0;
}

// ---- fast path: store exp() to slice-major scratch (coalesced) ----------
__global__ void energy_ws_kernel(const float* __restrict__ qp,
                                 const float* __restrict__ kp,
                                 const float* __restrict__ mask,
                                 float* __restrict__ eraw,   // [slice][q][k]
                                 float* __restrict__ colsum) {
    const int lane = threadIdx.x & 31;
    const int wid = blockIdx.x * (blockDim.x >> 5) + (threadIdx.x >> 5);
    int slice, q0, k0; float p[8];
    energy_tile(qp, kp, mask, colsum, wid, lane, &slice, &q0, &k0, p);
    const int half = lane >> 4, lm = lane & 15;
#pragma unroll
    for (int r = 0; r < 8; ++r) {
        const int qg = q0 + r + 8 * half;
        eraw[((size_t)slice * NN + qg) * NN + k0 + lm] = p[r];
    }
}

// ---- fallback path: store exp() straight into final (b,q,k,t,h) layout --
__global__ void energy_direct_kernel(const float* __restrict__ qp,
                                     const float* __restrict__ kp,
                                     const float* __restrict__ mask,
                                     float* __restrict__ att,
                                     float* __restrict__ colsum) {
    const int lane = threadIdx.x & 31;
    const int wid = blockIdx.x * (blockDim.x >> 5) + (threadIdx.x >> 5);
    int slice, q0, k0; float p[8];
    energy_tile(qp, kp, mask, colsum, wid, lane, &slice, &q0, &k0, p);
    const int half = lane >> 4, lm = lane & 15;
    const int h = slice % HH, t = (slice / HH) % TT, b = slice / SS;
#pragma unroll
    for (int r = 0; r < 8; ++r) {
        const int qg = q0 + r + 8 * half;
        att[((((size_t)b * NN + qg) * NN + k0 + lm) * TT + t) * HH + h] = p[r];
    }
}

// ---------------- fast path: TDM-fed LDS transpose + normalize -----------
// One block per (b,q). Final layout for fixed (b,q) is a contiguous
// 512*96-float run: att[base + k*96 + s], s = t*H+h, slice = b*96+s.
// The 96x64 source tile (row stride = N*N floats across slices) is pulled
// into LDS by the Tensor Data Mover; TDM's pad feature inserts 1 dword per
// 64 stored dwords -> LDS row stride 65 (bank-conflict free transpose).
#define TK 64                                    /* k chunk                */
#define TPAD 65                                  /* LDS row stride (pad)   */
#define TILE_F (SS * TPAD)                       /* floats per LDS tile    */

__device__ __forceinline__ void tdm_load_tile(const float* gsrc,
                                              unsigned int lds_off) {
    const unsigned long long ga = (unsigned long long)(size_t)gsrc;
    // D# group0: count=1 | lds_addr | global_addr[56:0] | type=2
    v4u g0 = { 1u,
               lds_off,
               (unsigned int)ga,
               (unsigned int)((ga >> 32) & 0x1FFFFFFull) | (2u << 30) };
    // D# group1: data_size=4B(2), pad_enable, pad_interval=64dw(5),
    //            pad_amount=1dw(0); tensor 64x96, tile 64x96,
    //            dim0_stride = N*N (slice pitch in elements)
    v8i g1 = { (int)((2u << 16) | (1u << 20) | (5u << 22)),
               (int)(TK << 16),            /* tensor_dim0 lo16            */
               (int)(SS << 16),            /* dim0 hi16=0 | tensor_dim1   */
               (int)(TK << 16),            /* dim1 hi16=0 | tile_dim0     */
               (int)SS,                    /* tile_dim1 | tile_dim2=0     */
               (int)(NN * NN),             /* tensor_dim0_stride lo32     */
               0,                          /* stride hi16 | dim1_stride   */
               0 };
    v4i gz4 = { 0, 0, 0, 0 };              /* 2-D tensor: groups 2/3 zero */
    v8i gz8 = { 0, 0, 0, 0, 0, 0, 0, 0 };  /* extra group (6-arg form)    */
    __builtin_amdgcn_tensor_load_to_lds(g0, g1, gz4, gz4, gz8, 0);
}

__global__ void trans_norm_tdm_kernel(const float* __restrict__ eraw,
                                      const float* __restrict__ colsum,
                                      float* __restrict__ att) {
    __shared__ float tiles[2][TILE_F];           // double-buffered TDM dest
    __shared__ float csl[TILE_F];                // colsum chunk (padded)
    const int tid = threadIdx.x;                 // 256 threads
    const int q = blockIdx.x & (NN - 1);
    const int b = blockIdx.x >> 9;
    const size_t obase = ((size_t)b * NN + q) * NN * SS;
    const float* gbase = eraw + ((size_t)(b * SS) * NN + q) * NN;

    int cur = 0;
    if (tid < 32) {                              // wave 0 drives the TDM
        tdm_load_tile(gbase, (unsigned int)(size_t)(void*)&tiles[0][0]);
    }

    for (int ic = 0; ic < NN / TK; ++ic) {
        const int kc = ic * TK;
        // stage colsum chunk (coalesced 256B rows) while TDM streams tile
#pragma unroll
        for (int i = 0; i < (SS * TK) / 256; ++i) {
            const int idx = tid + i * 256;
            const int s = idx / TK, kl = idx - s * TK;
            csl[s * TPAD + kl] =
                colsum[(size_t)(b * SS + s) * NN + kc + kl];
        }
        if (tid < 32) {
            if (ic < NN / TK - 1) {              // prefetch next chunk
                tdm_load_tile(gbase + kc + TK,
                              (unsigned int)(size_t)(void*)&tiles[cur ^ 1][0]);
                __builtin_amdgcn_s_wait_tensorcnt(1);  // current tile ready
            } else {
                __builtin_amdgcn_s_wait_tensorcnt(0);
            }
        }
        __syncthreads();
        // write 64*96 contiguous floats as dense float4 stores
#pragma unroll
        for (int i = 0; i < (SS * TK) / 1024; ++i) {
            const int j = tid * 4 + i * 1024;
            float4 v;
            float* vp4 = (float*)&v;
#pragma unroll
            for (int u = 0; u < 4; ++u) {
                const int jj = j + u;
                const int kl = jj / SS, s = jj - kl * SS;
                vp4[u] = tiles[cur][s * TPAD + kl] / csl[s * TPAD + kl];
            }
            *(float4*)(att + obase + (size_t)kc * SS + j) = v;
        }
        __syncthreads();                         // tile/csl reuse guard
        cur ^= 1;
    }
}

// ---------------- fallback: elementwise normalize in place ---------------
__global__ void norm_kernel(float* __restrict__ att,
                            const float* __restrict__ colsum) {
    size_t i = (size_t)blockIdx.x * blockDim.x + threadIdx.x; // B*N*N*T*H exact
    size_t r = i;
    const int h = (int)(r % HH); r /= HH;
    const int t = (int)(r % TT); r /= TT;
    const int k = (int)(r % NN); r /= NN;
    r /= NN;
    const int b = (int)r;
    const int slice = (b * TT + t) * HH + h;
    att[i] = att[i] / colsum[(size_t)slice * NN + k];
}

// ---------------- out_pre = Attention @ V  (fast: from eraw) -------------
__global__ void av_ws_kernel(const float* __restrict__ eraw,
                             const float* __restrict__ colsum,
                             const float* __restrict__ vp,
                             float* __restrict__ outpre) {
    const int lane = threadIdx.x & 31;
    const int wid = blockIdx.x * (blockDim.x >> 5) + (threadIdx.x >> 5);
    const int qt = wid & 31;
    const int slice = wid >> 5;               // < 192, exact grid
    const int h = slice % HH, t = (slice / HH) % TT, b = slice / SS;
    const int half = lane >> 4, lm = lane & 15;

    const float* V = vp + (size_t)slice * NN * DD;
    const float* E = eraw + (size_t)slice * NN * NN;
    const float* CS = colsum + (size_t)slice * NN;
    const int q0 = qt * 16;

    v8f c = {};
#pragma unroll 4
    for (int kk = 0; kk < NN; kk += 4) {
        const int kq = kk + 2 * half;
        v2f a, bv;
        a.x  = E[(size_t)(q0 + lm) * NN + kq] / CS[kq];        // normalized A
        a.y  = E[(size_t)(q0 + lm) * NN + kq + 1] / CS[kq + 1];
        bv.x = V[kq * DD + lm];                // B[k][d] = V[k][d]
        bv.y = V[(kq + 1) * DD + lm];
        c = wmma4(a, bv, c);
    }
#pragma unroll
    for (int r = 0; r < 8; ++r) {
        const int qg = q0 + r + 8 * half;
        outpre[(((size_t)b * NN + qg) * TT + t) * EE + h * DD + lm] = c[r];
    }
}

// ---------------- out_pre = Attention @ V  (fallback: from final att) ----
__global__ void av_direct_kernel(const float* __restrict__ att,
                                 const float* __restrict__ vp,
                                 float* __restrict__ outpre) {
    const int lane = threadIdx.x & 31;
    const int wid = blockIdx.x * (blockDim.x >> 5) + (threadIdx.x >> 5);
    const int qt = wid & 31;
    const int slice = wid >> 5;
    const int h = slice % HH, t = (slice / HH) % TT, b = slice / SS;
    const int half = lane >> 4, lm = lane & 15;

    const float* V = vp + (size_t)slice * NN * DD;
    const int q0 = qt * 16;

    v8f c = {};
#pragma unroll 4
    for (int kk = 0; kk < NN; kk += 4) {
        const int kq = kk + 2 * half;
        v2f a, bv;
        const size_t abase =
            ((((size_t)b * NN + (q0 + lm)) * NN + kq) * TT + t) * HH + h;
        a.x  = att[abase];
        a.y  = att[abase + (size_t)TT * HH];
        bv.x = V[kq * DD + lm];
        bv.y = V[(kq + 1) * DD + lm];
        c = wmma4(a, bv, c);
    }
#pragma unroll
    for (int r = 0; r < 8; ++r) {
        const int qg = q0 + r + 8 * half;
        outpre[(((size_t)b * NN + qg) * TT + t) * EE + h * DD + lm] = c[r];
    }
}

// ---------------- out = out_pre @ Wo^T + bo ------------------------------
__global__ void outproj_kernel(const float* __restrict__ outpre,
                               const float* __restrict__ Wo,
                               const float* __restrict__ bo,
                               float* __restrict__ out) {
    const int lane = threadIdx.x & 31;
    const int wid = blockIdx.x * (blockDim.x >> 5) + (threadIdx.x >> 5);
    const int ct = wid & 7;
    const int rt = wid >> 3;                  // < 768, exact grid
    const int half = lane >> 4, lm = lane & 15;
    const int r0 = rt * 16, n0 = ct * 16;

    v8f c;
    const float bias = bo[n0 + lm];           // bias depends on column only
#pragma unroll
    for (int r = 0; r < 8; ++r) c[r] = bias;

#pragma unroll 8
    for (int kk = 0; kk < EE; kk += 4) {
        const int d0 = kk + 2 * half;
        v2f a, bw;
        a.x  = outpre[(size_t)(r0 + lm) * EE + d0];
        a.y  = outpre[(size_t)(r0 + lm) * EE + d0 + 1];
        bw.x = Wo[(size_t)(n0 + lm) * EE + d0];   // B[k][n] = Wo[n][k]
        bw.y = Wo[(size_t)(n0 + lm) * EE + d0 + 1];
        c = wmma4(a, bw, c);
    }
#pragma unroll
    for (int r = 0; r < 8; ++r) {
        out[(size_t)(r0 + r + 8 * half) * EE + n0 + lm] = c[r];
    }
}

// ---------------- host glue ----------------------------------------------
extern "C" void kernel_launch(void* const* d_in, const int* in_sizes, int n_in,
                              void* d_out, int out_size, void* d_ws, size_t ws_size,
                              hipStream_t stream) {
    const float* values = (const float*)d_in[0];
    const float* keys   = (const float*)d_in[1];
    const float* query  = (const float*)d_in[2];
    const float* mask   = (const float*)d_in[3];
    const float* Wv     = (const float*)d_in[4];
    const float* Wk     = (const float*)d_in[5];
    const float* Wq     = (const float*)d_in[6];
    const float* Wo     = (const float*)d_in[7];
    const float* bo     = (const float*)d_in[8];
    float* out = (float*)d_out;

    const size_t PROJ = (size_t)SLICES * NN * DD;            // 1,572,864 floats
    const size_t OUT_ELEMS = (size_t)BB * NN * TT * EE;      // 1,572,864 floats
    const size_t ATT_ELEMS = (size_t)BB * NN * NN * TT * HH; // 50,331,648
    const size_t CS_ELEMS = (size_t)SLICES * NN;

    float* ws = (float*)d_ws;
    float* qp     = ws;
    float* kp     = qp + PROJ;
    float* vp     = kp + PROJ;
    float* colsum = vp + PROJ;
    float* outpre = colsum + CS_ELEMS;
    float* eraw   = outpre + OUT_ELEMS;      // fast path only (201 MB)

    float* att = out + OUT_ELEMS;            // second tuple element

    const size_t base_ws  = (3 * PROJ + CS_ELEMS + OUT_ELEMS) * sizeof(float);
    const size_t fast_ws  = base_ws + ATT_ELEMS * sizeof(float);
    const bool fast = (ws_size >= fast_ws);  // ws_size fixed -> deterministic

    zero_colsum_kernel<<<(SLICES * NN) / 256, 256, 0, stream>>>(colsum);
    proj_kernel<<<(unsigned)((3 * PROJ) / 256), 256, 0, stream>>>(
        values, keys, query, Wv, Wk, Wq, qp, kp, vp);

    if (fast) {
        // coalesced pipeline: eraw (slice-major) -> AV + TDM LDS-transpose
        energy_ws_kernel<<<(SLICES * 32 * 32) / 8, 256, 0, stream>>>(
            qp, kp, mask, eraw, colsum);
        av_ws_kernel<<<(SLICES * 32) / 8, 256, 0, stream>>>(
            eraw, colsum, vp, outpre);
        trans_norm_tdm_kernel<<<BB * NN, 256, 0, stream>>>(eraw, colsum, att);
    } else {
        // low-workspace fallback: direct scattered writes
        energy_direct_kernel<<<(SLICES * 32 * 32) / 8, 256, 0, stream>>>(
            qp, kp, mask, att, colsum);
        norm_kernel<<<(unsigned)(ATT_ELEMS / 256), 256, 0, stream>>>(att, colsum);
        av_direct_kernel<<<(SLICES * 32) / 8, 256, 0, stream>>>(att, vp, outpre);
    }

    outproj_kernel<<<(768 * 8) / 8, 256, 0, stream>>>(outpre, Wo, bo, out);
}